// get_loss_60129542194
// MI455X (gfx1250) — compile-verified
//
#include <hip/hip_runtime.h>

typedef float v2f   __attribute__((ext_vector_type(2)));
typedef float v8f   __attribute__((ext_vector_type(8)));
typedef float f32x4 __attribute__((ext_vector_type(4)));

#define NUM_JOINTS 21
#define BLOCK 256
#define WAVES_PER_BLOCK (BLOCK / 32)
#define GRID_BLOCKS 2688   // 2688*256 threads * 8 chunks = 21*2^18 chunks exactly

// Full-wave (32-lane) sum via the matrix pipe:
// A(16x4): lane L<16 holds A[L][0]=v, lane L>=16 holds A[L-16][2]=v, K=1,3 are 0.
// B(4x16) = ones  =>  D[m][n] = v_m + v_{m+16}.
// Lane layout of D: lanes 0-15 hold M=0..7 in VGPRs 0..7, lanes 16-31 hold M=8..15.
// Summing a lane's 8 D values gives the half-sum; xor-16 shuffle joins the halves.
__device__ __forceinline__ float wave_reduce_wmma(float v) {
    v2f a; a.x = v;    a.y = 0.0f;
    v2f b; b.x = 1.0f; b.y = 1.0f;
    v8f c = {};
    c = __builtin_amdgcn_wmma_f32_16x16x4_f32(
        /*neg_a=*/false, a, /*neg_b=*/false, b,
        /*c_mod=*/(short)0, c, /*reuse_a=*/false, /*reuse_b=*/false);
    float t = c[0] + c[1] + c[2] + c[3] + c[4] + c[5] + c[6] + c[7];
    t += __shfl_xor(t, 16, 32);
    return t;  // full 32-lane sum, broadcast to every lane
}

__device__ __forceinline__ float joint_dist(float dx, float dy, float dz) {
    return sqrtf(fmaf(dx, dx, fmaf(dy, dy, dz * dz)));
}

// Stage 1: stream pred/target, accumulate per-thread sums of per-joint L2
// distances, reduce to one partial per block.
__global__ void __launch_bounds__(BLOCK)
joint_loss_partial(const float* __restrict__ pred,
                   const float* __restrict__ target,
                   float* __restrict__ partials,
                   long long nChunks,      // chunks of 4 joints (12 floats)
                   long long nJoints) {    // total joints (for remainder)
    const long long tid    = (long long)blockIdx.x * BLOCK + threadIdx.x;
    const long long stride = (long long)gridDim.x * BLOCK;

    float acc = 0.0f;
    for (long long i = tid; i < nChunks; i += stride) {
        const long long base = i * 12;  // 48B-aligned chunk
        const f32x4* p = (const f32x4*)(pred + base);
        const f32x4* t = (const f32x4*)(target + base);
        f32x4 p0 = __builtin_nontemporal_load(p);
        f32x4 p1 = __builtin_nontemporal_load(p + 1);
        f32x4 p2 = __builtin_nontemporal_load(p + 2);
        f32x4 t0 = __builtin_nontemporal_load(t);
        f32x4 t1 = __builtin_nontemporal_load(t + 1);
        f32x4 t2 = __builtin_nontemporal_load(t + 2);
        acc += joint_dist(p0.x - t0.x, p0.y - t0.y, p0.z - t0.z);
        acc += joint_dist(p0.w - t0.w, p1.x - t1.x, p1.y - t1.y);
        acc += joint_dist(p1.z - t1.z, p1.w - t1.w, p2.x - t2.x);
        acc += joint_dist(p2.y - t2.y, p2.z - t2.z, p2.w - t2.w);
    }

    // Remainder joints (none for B=2^20, but keep it shape-generic).
    if (tid == 0) {
        for (long long j = nChunks * 4; j < nJoints; ++j) {
            const long long f = j * 3;
            acc += joint_dist(pred[f] - target[f], pred[f + 1] - target[f + 1],
                              pred[f + 2] - target[f + 2]);
        }
    }

    // wave -> block reduction
    float wsum = wave_reduce_wmma(acc);
    __shared__ float lds[WAVES_PER_BLOCK];
    const int lane = threadIdx.x & 31;
    const int wave = threadIdx.x >> 5;
    if (lane == 0) lds[wave] = wsum;
    __syncthreads();
    if (wave == 0) {
        float s = (lane < WAVES_PER_BLOCK) ? lds[lane] : 0.0f;
        s = wave_reduce_wmma(s);
        if (lane == 0) partials[blockIdx.x] = s;
    }
}

// Stage 2: one block deterministically folds the per-block partials and scales.
__global__ void __launch_bounds__(BLOCK)
joint_loss_finalize(const float* __restrict__ partials,
                    float* __restrict__ out,
                    int n, float invCount) {
    float acc = 0.0f;
    for (int i = threadIdx.x; i < n; i += BLOCK) acc += partials[i];

    float wsum = wave_reduce_wmma(acc);
    __shared__ float lds[WAVES_PER_BLOCK];
    const int lane = threadIdx.x & 31;
    const int wave = threadIdx.x >> 5;
    if (lane == 0) lds[wave] = wsum;
    __syncthreads();
    if (wave == 0) {
        float s = (lane < WAVES_PER_BLOCK) ? lds[lane] : 0.0f;
        s = wave_reduce_wmma(s);
        if (lane == 0) out[0] = s * invCount;
    }
}

extern "C" void kernel_launch(void* const* d_in, const int* in_sizes, int n_in,
                              void* d_out, int out_size, void* d_ws, size_t ws_size,
                              hipStream_t stream) {
    const float* pred   = (const float*)d_in[0];
    const float* target = (const float*)d_in[1];
    float* out      = (float*)d_out;
    float* partials = (float*)d_ws;  // GRID_BLOCKS floats << ws_size

    const long long B       = (long long)in_sizes[0] / (NUM_JOINTS * 3);
    const long long nJoints = B * NUM_JOINTS;
    const long long nChunks = nJoints / 4;  // 4 joints = 12 floats per chunk

    joint_loss_partial<<<GRID_BLOCKS, BLOCK, 0, stream>>>(
        pred, target, partials, nChunks, nJoints);

    const float invCount = (float)(1.0 / (double)nJoints);
    joint_loss_finalize<<<1, BLOCK, 0, stream>>>(partials, out, GRID_BLOCKS, invCount);
}